// FixedAttention_61933428416271
// MI455X (gfx1250) — compile-verified
//
#include <hip/hip_runtime.h>

// Problem sizes (fixed by the reference)
#define B_ 16
#define L_ 2048
#define E_ 256
#define BE_ (B_ * E_)
#define ROWS_PER_BLOCK 64

typedef __attribute__((ext_vector_type(2))) float v2f;
typedef __attribute__((ext_vector_type(8))) float v8f;

// ---------------------------------------------------------------------------
// Kernel 0: zero the workspace accumulators (xsum, t, u, c, Wsum)
// ---------------------------------------------------------------------------
__global__ void k_zero(float* __restrict__ ws, int n) {
    int i = blockIdx.x * blockDim.x + threadIdx.x;
    if (i < n) ws[i] = 0.0f;
}

// ---------------------------------------------------------------------------
// Kernel 1: Xsum[b,e] = sum_{l < len_b} x[b,l,e]   (streaming pass #1 over x)
// grid (B, 16 chunks), 256 threads (thread == e column) -> coalesced rows
// ---------------------------------------------------------------------------
__global__ void k_rowsum(const float* __restrict__ x,
                         const int* __restrict__ lengths,
                         float* __restrict__ xsum) {
    const int b = blockIdx.x;
    const int chunk = blockIdx.y;
    const int e = threadIdx.x;
    const int len = lengths[b];
    const int r0 = chunk * (L_ / 16);
    int r1 = r0 + (L_ / 16);
    if (r1 > len) r1 = len;
    if (r0 >= r1) return;
    const float* xb = x + (size_t)b * L_ * E_;
    float acc = 0.0f;
    for (int l = r0; l < r1; ++l) acc += xb[(size_t)l * E_ + e];
    atomicAdd(&xsum[b * E_ + e], acc);
}

// ---------------------------------------------------------------------------
// Kernel 2 (WMMA): s = Xsum @ Wq^T + len*bq   (kept in LDS),
//                  c_b = s_b . bk,   t = s @ Wk    -> workspace
// One workgroup, 512 threads = 16 waves; wave w owns N-tile [16w, 16w+16).
// Uses V_WMMA_F32_16X16X4_F32: M=16 (batch), K steps of 4 over E=256.
// ---------------------------------------------------------------------------
__global__ void k_setup(const float* __restrict__ xsum,
                        const int* __restrict__ lengths,
                        const float* __restrict__ Wq, const float* __restrict__ bq,
                        const float* __restrict__ Wk, const float* __restrict__ bk,
                        float* __restrict__ t, float* __restrict__ cvec) {
    __shared__ float s_lds[BE_];   // 16 KB: s matrix [B=16, E=256]
    const int tid   = threadIdx.x;
    const int wave  = tid >> 5;
    const int lane  = tid & 31;
    const int laneM = lane & 15;   // M (lanes 0-15) / M again (lanes 16-31)
    const int hi    = lane >> 4;   // upper half holds K+2,K+3 (A/B), M+8 (C/D)

    // ---- GEMM 1: s[b,f] = sum_e Xsum[b,e] * Wq[f,e]  (+ len_b * bq[f]) ----
    {
        const int fbase = wave * 16 + laneM;
        v8f acc = {};
        for (int kk = 0; kk < E_ / 4; ++kk) {
            const int k0 = kk * 4 + hi * 2;
            v2f a, bf;
            a.x  = xsum[laneM * E_ + k0];
            a.y  = xsum[laneM * E_ + k0 + 1];
            bf.x = Wq[fbase * E_ + k0];          // B[k, n=f] = Wq[f*E + k]
            bf.y = Wq[fbase * E_ + k0 + 1];
            acc = __builtin_amdgcn_wmma_f32_16x16x4_f32(
                false, a, false, bf, (short)0, acc, false, false);
        }
#pragma unroll
        for (int r = 0; r < 8; ++r) {
            const int bm = r + hi * 8;           // C/D: lanes16-31 hold M=8..15
            s_lds[bm * E_ + fbase] = acc[r] + (float)lengths[bm] * bq[fbase];
        }
    }
    __syncthreads();

    // ---- c_b = s_b . bk  (tiny: 16 serial dot products) ----
    if (tid < B_) {
        float cc = 0.0f;
        for (int f = 0; f < E_; ++f) cc += s_lds[tid * E_ + f] * bk[f];
        cvec[tid] = cc;
    }

    // ---- GEMM 2: t[b,e] = sum_f s[b,f] * Wk[f,e] ----
    {
        const int ebase = wave * 16 + laneM;
        v8f acc = {};
        for (int kk = 0; kk < E_ / 4; ++kk) {
            const int k0 = kk * 4 + hi * 2;
            v2f a, bf;
            a.x  = s_lds[laneM * E_ + k0];
            a.y  = s_lds[laneM * E_ + k0 + 1];
            bf.x = Wk[k0 * E_ + ebase];          // B[k, n=e] = Wk[k*E + e]
            bf.y = Wk[(k0 + 1) * E_ + ebase];
            acc = __builtin_amdgcn_wmma_f32_16x16x4_f32(
                false, a, false, bf, (short)0, acc, false, false);
        }
#pragma unroll
        for (int r = 0; r < 8; ++r) {
            const int bm = r + hi * 8;
            t[bm * E_ + ebase] = acc[r];
        }
    }
}

// ---------------------------------------------------------------------------
// Kernel 3: streaming pass #2 over x.
//   w = t_b . x[b,l,:] + c_b    (only l < len_b)
//   u_b += w * x[b,l,:] ;  W_b += w
// grid (B, L/64), 256 threads = 8 waves, wave owns one row at a time:
// lane reads e = lane + 32m (coalesced), butterfly-reduce dot, reuse regs
// for the rank-1 update. LDS reduce across waves, then global f32 atomics.
// ---------------------------------------------------------------------------
__global__ void k_pass2(const float* __restrict__ x,
                        const int* __restrict__ lengths,
                        const float* __restrict__ t, const float* __restrict__ cvec,
                        float* __restrict__ u, float* __restrict__ Wsum) {
    const int b     = blockIdx.x;
    const int chunk = blockIdx.y;
    const int tid   = threadIdx.x;   // 256
    const int wave  = tid >> 5;      // 8 waves
    const int lane  = tid & 31;

    __shared__ float t_lds[E_];
    __shared__ float u_lds[E_];
    __shared__ float w_lds;
    t_lds[tid] = t[b * E_ + tid];
    u_lds[tid] = 0.0f;
    if (tid == 0) w_lds = 0.0f;
    __syncthreads();

    const int len = lengths[b];
    const float cb = cvec[b];
    const float* xb = x + (size_t)b * L_ * E_;
    const int r0 = chunk * ROWS_PER_BLOCK;

    float uacc[8];
#pragma unroll
    for (int m = 0; m < 8; ++m) uacc[m] = 0.0f;
    float wtot = 0.0f;

    for (int it = 0; it < ROWS_PER_BLOCK / 8; ++it) {
        const int l = r0 + it * 8 + wave;        // wave-uniform row index
        if (l < len) {
            const float* row = xb + (size_t)l * E_;
            __builtin_prefetch(row + 8 * E_, 0, 0);  // gfx1250 global_prefetch_b8
            float xv[8];
            float p = 0.0f;
#pragma unroll
            for (int m = 0; m < 8; ++m) {
                xv[m] = row[lane + 32 * m];
                p += xv[m] * t_lds[lane + 32 * m];
            }
#pragma unroll
            for (int off = 16; off >= 1; off >>= 1)
                p += __shfl_xor(p, off, 32);      // all lanes get full dot
            const float w = p + cb;
            wtot += w;
#pragma unroll
            for (int m = 0; m < 8; ++m) uacc[m] += w * xv[m];
        }
    }
#pragma unroll
    for (int m = 0; m < 8; ++m) atomicAdd(&u_lds[lane + 32 * m], uacc[m]);
    if (lane == 0) atomicAdd(&w_lds, wtot);
    __syncthreads();

    atomicAdd(&u[b * E_ + tid], u_lds[tid]);
    if (tid == 0) atomicAdd(&Wsum[b], w_lds);
}

// ---------------------------------------------------------------------------
// Kernel 4 (WMMA): out[b,e] = (u_b . Wv[e,:] + W_b * bv[e]) / L
// Same single-workgroup WMMA pattern as kernel 2's GEMM 1.
// ---------------------------------------------------------------------------
__global__ void k_out(const float* __restrict__ u, const float* __restrict__ Wsum,
                      const float* __restrict__ Wv, const float* __restrict__ bv,
                      float* __restrict__ out) {
    const int tid   = threadIdx.x;   // 512
    const int wave  = tid >> 5;
    const int lane  = tid & 31;
    const int laneM = lane & 15;
    const int hi    = lane >> 4;
    const int ebase = wave * 16 + laneM;

    v8f acc = {};
    for (int kk = 0; kk < E_ / 4; ++kk) {
        const int k0 = kk * 4 + hi * 2;
        v2f a, bf;
        a.x  = u[laneM * E_ + k0];
        a.y  = u[laneM * E_ + k0 + 1];
        bf.x = Wv[ebase * E_ + k0];              // B[k, n=e] = Wv[e*E + k]
        bf.y = Wv[ebase * E_ + k0 + 1];
        acc = __builtin_amdgcn_wmma_f32_16x16x4_f32(
            false, a, false, bf, (short)0, acc, false, false);
    }
    const float invL = 1.0f / (float)L_;
#pragma unroll
    for (int r = 0; r < 8; ++r) {
        const int bm = r + hi * 8;
        out[bm * E_ + ebase] = (acc[r] + Wsum[bm] * bv[ebase]) * invL;
    }
}

// ---------------------------------------------------------------------------
// Launch
// ---------------------------------------------------------------------------
extern "C" void kernel_launch(void* const* d_in, const int* in_sizes, int n_in,
                              void* d_out, int out_size, void* d_ws, size_t ws_size,
                              hipStream_t stream) {
    const float* x       = (const float*)d_in[0];
    const int*   lengths = (const int*)d_in[1];
    const float* Wq      = (const float*)d_in[2];
    const float* bq      = (const float*)d_in[3];
    const float* Wk      = (const float*)d_in[4];
    const float* bk      = (const float*)d_in[5];
    const float* Wv      = (const float*)d_in[6];
    const float* bv      = (const float*)d_in[7];
    float*       out     = (float*)d_out;
    float*       ws      = (float*)d_ws;

    // workspace layout (floats): xsum[BE] | t[BE] | u[BE] | c[B] | Wsum[B]
    float* xsum = ws;
    float* t    = ws + BE_;
    float* u    = ws + 2 * BE_;
    float* cvec = ws + 3 * BE_;
    float* Wsum = ws + 3 * BE_ + B_;
    const int nzero = 3 * BE_ + 2 * B_;

    k_zero<<<(nzero + 255) / 256, 256, 0, stream>>>(ws, nzero);
    k_rowsum<<<dim3(B_, 16), 256, 0, stream>>>(x, lengths, xsum);
    k_setup<<<1, 512, 0, stream>>>(xsum, lengths, Wq, bq, Wk, bk, t, cvec);
    k_pass2<<<dim3(B_, L_ / ROWS_PER_BLOCK), 256, 0, stream>>>(x, lengths, t, cvec, u, Wsum);
    k_out<<<1, 512, 0, stream>>>(u, Wsum, Wv, bv, out);
}